// BipartiteLayer_29892972380756
// MI455X (gfx1250) — compile-verified
//
#include <hip/hip_runtime.h>
#include <hip/hip_bf16.h>

typedef __attribute__((ext_vector_type(2))) float v2f;
typedef __attribute__((ext_vector_type(8))) float v8f;

// ---------------------------------------------------------------------------
// Phase 1: xp[M,256] = x[M,128] @ W[128,256] + b        (fp32 WMMA 16x16x4)
// One wave per 16x16 output tile. 8 waves / 256-thread block.
// A-frag: lane l -> row m = l&15, K pair kk = 2*(l>>4): {k0+kk, k0+kk+1}
// B-frag: lane l -> col n = l&15, same K pair.
// D: VGPR v, lane l -> (m = v + 8*(l>>4), n = l&15)
// ---------------------------------------------------------------------------
__global__ __launch_bounds__(256) void input_proj_kernel(
    const float* __restrict__ x, const float* __restrict__ W,
    const float* __restrict__ bias, float* __restrict__ xp, int M)
{
    const int lane = threadIdx.x & 31;
    const int wave = threadIdx.x >> 5;
    const int tilesM = (M + 15) >> 4;
    const int tile = blockIdx.x * 8 + wave;
    if (tile >= tilesM * 16) return;                 // wave-uniform exit
    const int tm = tile >> 4;                        // M tile
    const int tn = tile & 15;                        // N tile (256/16)
    int m = tm * 16 + (lane & 15);
    if (m >= M) m = M - 1;                           // clamp reads (tail-safe)
    const int n  = tn * 16 + (lane & 15);
    const int kk = (lane >> 4) * 2;

    const float* __restrict__ arow = x + (size_t)m * 128;
    v8f c = {};
    for (int k0 = 0; k0 < 128; k0 += 32) {
        __builtin_prefetch(arow + k0 + 32, 0, 1);    // global_prefetch next chunk
        #pragma unroll
        for (int k = k0; k < k0 + 32; k += 4) {
            v2f a = *(const v2f*)(arow + k + kk);
            v2f b;
            b.x = W[(size_t)(k + kk)     * 256 + n];
            b.y = W[(size_t)(k + kk + 1) * 256 + n];
            c = __builtin_amdgcn_wmma_f32_16x16x4_f32(false, a, false, b,
                                                      (short)0, c, false, false);
        }
    }
    const float bn = bias[n];
    const int mhi = (lane >> 4) * 8;
    float* __restrict__ orow = xp + (size_t)(tm * 16 + mhi) * 256 + n;
    if (tm * 16 + 16 <= M) {                         // wave-uniform full tile
        #pragma unroll
        for (int v = 0; v < 8; ++v) orow[(size_t)v * 256] = c[v] + bn;
    } else {
        #pragma unroll
        for (int v = 0; v < 8; ++v) {
            const int mm = tm * 16 + mhi + v;
            if (mm < M) xp[(size_t)mm * 256 + n] = c[v] + bn;
        }
    }
}

// ---------------------------------------------------------------------------
// Phase 2: per-edge score + scatter sum/count/max via atomics.
// One wave per edge; lane l owns float4 chunks at feature dims
//   {4l..4l+3, 128+4l.., 256+4l.., 384+4l..}  (512 dims total, coalesced).
// max buffers start at 0; reference clamps max at 0, so only v>0 matters and
// positive-float bit patterns are monotone as ints -> int atomicMax is exact.
// ---------------------------------------------------------------------------
__device__ __forceinline__ void atomic_max_pos(float* addr, float v) {
    if (v > 0.0f) atomicMax((int*)addr, __float_as_int(v));
}

__global__ __launch_bounds__(256) void edge_kernel(
    const float* __restrict__ xp_intt, const float* __restrict__ xp_mvtx,
    const int* __restrict__ eidx,            // [2,E]: start | end
    const float* __restrict__ Wsc, const float* __restrict__ bsc,
    float* __restrict__ sum_i, float* __restrict__ max_i, float* __restrict__ cnt_i,
    float* __restrict__ sum_m, float* __restrict__ max_m, float* __restrict__ cnt_m,
    int E)
{
    const int lane = threadIdx.x & 31;
    const int e = blockIdx.x * 8 + (threadIdx.x >> 5);
    if (e >= E) return;                       // wave-uniform
    const int s = eidx[e];
    const int d = eidx[E + e];

    const float4* ra = (const float4*)(xp_intt + (size_t)s * 256);
    const float4* rb = (const float4*)(xp_mvtx + (size_t)d * 256);
    const float4* wv = (const float4*)Wsc;    // 128 float4s

    float4 a0 = ra[lane], a1 = ra[lane + 32];
    float4 b0 = rb[lane], b1 = rb[lane + 32];
    float4 w0 = wv[lane], w1 = wv[lane + 32], w2 = wv[lane + 64], w3 = wv[lane + 96];

    float p = a0.x*w0.x + a0.y*w0.y + a0.z*w0.z + a0.w*w0.w
            + a1.x*w1.x + a1.y*w1.y + a1.z*w1.z + a1.w*w1.w
            + b0.x*w2.x + b0.y*w2.y + b0.z*w2.z + b0.w*w2.w
            + b1.x*w3.x + b1.y*w3.y + b1.z*w3.z + b1.w*w3.w;
    #pragma unroll
    for (int off = 16; off > 0; off >>= 1) p += __shfl_xor(p, off, 32);

    const float sc = __expf(-fabsf(p + bsc[0]));

    float v[16] = { a0.x*sc, a0.y*sc, a0.z*sc, a0.w*sc,
                    a1.x*sc, a1.y*sc, a1.z*sc, a1.w*sc,
                    b0.x*sc, b0.y*sc, b0.z*sc, b0.w*sc,
                    b1.x*sc, b1.y*sc, b1.z*sc, b1.w*sc };
    const int base[4] = { 4*lane, 128 + 4*lane, 256 + 4*lane, 384 + 4*lane };

    float* si = sum_i + (size_t)s * 512;
    float* mi = max_i + (size_t)s * 512;
    float* sm = sum_m + (size_t)d * 512;
    float* mm = max_m + (size_t)d * 512;
    #pragma unroll
    for (int ch = 0; ch < 4; ++ch) {
        #pragma unroll
        for (int j = 0; j < 4; ++j) {
            const int idx = base[ch] + j;
            const float val = v[ch * 4 + j];
            atomicAdd(si + idx, val);
            atomicAdd(sm + idx, val);
            atomic_max_pos(mi + idx, val);
            atomic_max_pos(mm + idx, val);
        }
    }
    if (lane == 0) {
        atomicAdd(cnt_i + s, 1.0f);
        atomicAdd(cnt_m + d, 1.0f);
    }
}

// ---------------------------------------------------------------------------
// Phase 3: h[M,128] = relu(H[M,1408] @ W[1408,128] + b), H fused from
// {x(128), xp(256), sum*inv(512), max(512)}. One wave per 16x16 tile,
// block = 8 waves covering all 8 N tiles of one 16-row M stripe.
// WMMA_SEG runs one K segment: A from `src` (optionally scaled), B from
// W rows [wbase + k].
// ---------------------------------------------------------------------------
__global__ __launch_bounds__(256) void output_proj_kernel(
    const float* __restrict__ x,    const float* __restrict__ xp,
    const float* __restrict__ sums, const float* __restrict__ maxs,
    const float* __restrict__ cnt,  const float* __restrict__ W,
    const float* __restrict__ bias, float* __restrict__ out, int M)
{
    const int lane = threadIdx.x & 31;
    const int wave = threadIdx.x >> 5;        // N tile 0..7
    const int tm   = blockIdx.x;              // M tile
    int m = tm * 16 + (lane & 15);
    if (m >= M) m = M - 1;
    const int n  = wave * 16 + (lane & 15);
    const int kk = (lane >> 4) * 2;

    const float inv = 1.0f / fmaxf(cnt[m], 1.0f);
    const float* __restrict__ xr  = x    + (size_t)m * 128;
    const float* __restrict__ xpr = xp   + (size_t)m * 256;
    const float* __restrict__ sr  = sums + (size_t)m * 512;
    const float* __restrict__ mr  = maxs + (size_t)m * 512;

    v8f c = {};

#define WMMA_SEG(src, len, wbase, scale)                                       \
    for (int k0 = 0; k0 < (len); k0 += 32) {                                   \
        __builtin_prefetch((src) + k0 + 32, 0, 1);                             \
        _Pragma("unroll")                                                      \
        for (int k = k0; k < k0 + 32; k += 4) {                                \
            v2f a = *(const v2f*)((src) + k + kk);                             \
            a.x *= (scale); a.y *= (scale);                                    \
            v2f b;                                                             \
            b.x = W[(size_t)((wbase) + k + kk)     * 128 + n];                 \
            b.y = W[(size_t)((wbase) + k + kk + 1) * 128 + n];                 \
            c = __builtin_amdgcn_wmma_f32_16x16x4_f32(false, a, false, b,      \
                                                      (short)0, c, false, false);\
        }                                                                      \
    }

    WMMA_SEG(xr,  128,   0, 1.0f)   // raw input x
    WMMA_SEG(xpr, 256, 128, 1.0f)   // projected xp
    WMMA_SEG(sr,  512, 384, inv)    // mean = sum * 1/max(cnt,1)
    WMMA_SEG(mr,  512, 896, 1.0f)   // max pool
#undef WMMA_SEG

    const float bn = bias[n];
    const int mhi = (lane >> 4) * 8;
    float* __restrict__ orow = out + (size_t)(tm * 16 + mhi) * 128 + n;
    if (tm * 16 + 16 <= M) {                  // wave-uniform full tile
        #pragma unroll
        for (int v = 0; v < 8; ++v)
            orow[(size_t)v * 128] = fmaxf(c[v] + bn, 0.0f);
    } else {
        #pragma unroll
        for (int v = 0; v < 8; ++v) {
            const int mm = tm * 16 + mhi + v;
            if (mm < M) out[(size_t)mm * 128 + n] = fmaxf(c[v] + bn, 0.0f);
        }
    }
}

// ---------------------------------------------------------------------------
extern "C" void kernel_launch(void* const* d_in, const int* in_sizes, int n_in,
                              void* d_out, int out_size, void* d_ws, size_t ws_size,
                              hipStream_t stream) {
    const float* x_intt   = (const float*)d_in[0];
    const float* x_mvtx   = (const float*)d_in[1];
    const int*   eidx     = (const int*)  d_in[2];
    const float* W_in_i   = (const float*)d_in[3];
    const float* b_in_i   = (const float*)d_in[4];
    const float* W_in_m   = (const float*)d_in[5];
    const float* b_in_m   = (const float*)d_in[6];
    const float* W_score  = (const float*)d_in[7];
    const float* b_score  = (const float*)d_in[8];
    const float* W_out_i  = (const float*)d_in[9];
    const float* b_out_i  = (const float*)d_in[10];
    const float* W_out_m  = (const float*)d_in[11];
    const float* b_out_m  = (const float*)d_in[12];

    const int Mi = in_sizes[0] / 128;   // 50000
    const int Mm = in_sizes[1] / 128;   // 50000
    const int E  = in_sizes[2] / 2;     // 320000

    // Workspace layout (fp32):
    float* xp_i  = (float*)d_ws;
    float* xp_m  = xp_i  + (size_t)Mi * 256;
    float* sum_i = xp_m  + (size_t)Mm * 256;
    float* sum_m = sum_i + (size_t)Mi * 512;
    float* max_i = sum_m + (size_t)Mm * 512;
    float* max_m = max_i + (size_t)Mi * 512;
    float* cnt_i = max_m + (size_t)Mm * 512;
    float* cnt_m = cnt_i + Mi;

    // Zero sums/maxes/counts (max init 0.0f == clamp-at-zero semantics).
    const size_t zbytes = ((size_t)Mi * 512 * 2 + (size_t)Mm * 512 * 2
                           + (size_t)Mi + (size_t)Mm) * sizeof(float);
    hipMemsetAsync(sum_i, 0, zbytes, stream);

    // Phase 1: input projections (16 N-tiles each, 8 waves/block).
    {
        const int tiles_i = ((Mi + 15) / 16) * 16;
        const int tiles_m = ((Mm + 15) / 16) * 16;
        input_proj_kernel<<<(tiles_i + 7) / 8, 256, 0, stream>>>(
            x_intt, W_in_i, b_in_i, xp_i, Mi);
        input_proj_kernel<<<(tiles_m + 7) / 8, 256, 0, stream>>>(
            x_mvtx, W_in_m, b_in_m, xp_m, Mm);
    }

    // Phase 2: edge score + scatter (1 wave/edge, 8 edges/block).
    edge_kernel<<<(E + 7) / 8, 256, 0, stream>>>(
        xp_i, xp_m, eidx, W_score, b_score,
        sum_i, max_i, cnt_i, sum_m, max_m, cnt_m, E);

    // Phase 3: fused H assembly + output GEMM + ReLU.
    float* h_intt = (float*)d_out;
    float* h_mvtx = h_intt + (size_t)Mi * 128;
    output_proj_kernel<<<(Mi + 15) / 16, 256, 0, stream>>>(
        x_intt, xp_i, sum_i, max_i, cnt_i, W_out_i, b_out_i, h_intt, Mi);
    output_proj_kernel<<<(Mm + 15) / 16, 256, 0, stream>>>(
        x_mvtx, xp_m, sum_m, max_m, cnt_m, W_out_m, b_out_m, h_mvtx, Mm);
}